// RecurrentRGCN_18176301597389
// MI455X (gfx1250) — compile-verified
//
#include <hip/hip_runtime.h>
#include <hip/hip_bf16.h>
#include <math.h>

// ---------------------------------------------------------------------------
// RecurrentRGCN on MI455X (gfx1250, wave32).
// Factored form:  segment_sum((h[src]+rel)@W) == segment_sum(h[src]+rel) @ W
// -> edge pass is gather + atomic scatter-add (L2-resident: 83MB tables,
//    192MB L2), matmuls shrink to N x 200 x 200 fp32 done with
//    V_WMMA_F32_16X16X4_F32 (exact fp32 path).
// ---------------------------------------------------------------------------

#define DIM   200
#define DP    208          // padded (13 tiles of 16)
#define SLOPE 0.2291666666666667f   // (1/8 + 1/3)/2

typedef __attribute__((ext_vector_type(2))) float v2f;
typedef __attribute__((ext_vector_type(8))) float v8f;

__device__ __forceinline__ float wave_sum32(float v) {
    #pragma unroll
    for (int off = 16; off > 0; off >>= 1) v += __shfl_xor(v, off, 32);
    return v;
}

// ---------------- utility kernels ----------------

__global__ void k_zero(float* __restrict__ p, long n) {
    long i = (long)blockIdx.x * blockDim.x + threadIdx.x;
    long stride = (long)gridDim.x * blockDim.x;
    for (; i < n; i += stride) p[i] = 0.0f;
}

__global__ void k_copy(const float* __restrict__ s, float* __restrict__ d, int n) {
    int i = blockIdx.x * blockDim.x + threadIdx.x;
    if (i < n) d[i] = s[i];
}

// pad 200x200 weight into 208x208 (zero padded)
__global__ void k_pad_weight(const float* __restrict__ W, float* __restrict__ Wp) {
    int i = blockIdx.x * blockDim.x + threadIdx.x;
    if (i >= DP * DP) return;
    int r = i / DP, c = i % DP;
    Wp[i] = (r < DIM && c < DIM) ? W[r * DIM + c] : 0.0f;
}

// in-degree histogram
__global__ void k_deg(const int* __restrict__ dst, float* __restrict__ deg, int E) {
    int e = blockIdx.x * blockDim.x + threadIdx.x;
    if (e < E) atomicAdd(&deg[dst[e]], 1.0f);
}

// h0 = l2norm(dynamic_emb) into padded layout
__global__ void k_init_h(const float* __restrict__ dyn, float* __restrict__ hP, int N) {
    int wave = threadIdx.x >> 5, lane = threadIdx.x & 31;
    int row = blockIdx.x * 8 + wave;
    if (row >= N) return;
    float vals[7]; float ss = 0.0f;
    #pragma unroll
    for (int j = 0; j < 7; ++j) {
        int i = lane + j * 32;
        float x = (i < DIM) ? dyn[(long)row * DIM + i] : 0.0f;
        vals[j] = x; ss += x * x;
    }
    ss = wave_sum32(ss);
    float inv = 1.0f / fmaxf(sqrtf(ss), 1e-12f);
    #pragma unroll
    for (int j = 0; j < 7; ++j) {
        int i = lane + j * 32;
        if (i < DP) hP[(long)row * DP + i] = vals[j] * inv;
    }
}

// in-place row l2norm of padded buffer
__global__ void k_l2norm_rows(float* __restrict__ p, int N) {
    int wave = threadIdx.x >> 5, lane = threadIdx.x & 31;
    int row = blockIdx.x * 8 + wave;
    if (row >= N) return;
    float vals[7]; float ss = 0.0f;
    #pragma unroll
    for (int j = 0; j < 7; ++j) {
        int i = lane + j * 32;
        float x = (i < DIM) ? p[(long)row * DP + i] : 0.0f;
        vals[j] = x; ss += x * x;
    }
    ss = wave_sum32(ss);
    float inv = 1.0f / fmaxf(sqrtf(ss), 1e-12f);
    #pragma unroll
    for (int j = 0; j < 7; ++j) {
        int i = lane + j * 32;
        if (i < DP) p[(long)row * DP + i] = (i < DIM) ? vals[j] * inv : 0.0f;
    }
}

// edge scatter: pre[dst] += in[src] + rel[etype]   (one wave per edge)
__global__ void k_scatter(const float* __restrict__ inP, const int* __restrict__ src,
                          const int* __restrict__ dst, const int* __restrict__ et,
                          const float* __restrict__ rel, float* __restrict__ preP, int E) {
    int wave = threadIdx.x >> 5, lane = threadIdx.x & 31;
    int e = blockIdx.x * 8 + wave;
    if (e >= E) return;
    int s = src[e], d = dst[e], r = et[e];
    const float* __restrict__ hs = inP + (long)s * DP;
    const float* __restrict__ rr = rel + (long)r * DIM;
    float* __restrict__ pd = preP + (long)d * DP;
    for (int i = lane; i < DIM; i += 32)
        atomicAdd(&pd[i], hs[i] + rr[i]);
}

// ---------------- WMMA matmul: out = act(A @ Wp [, *1/max(deg,1)] [, +bias]) ----
// mode 0: scale by 1/max(deg,1) then rrelu (eval)      (deg != nullptr)
// mode 1: add bias then sigmoid                        (bias != nullptr)
__global__ void k_mm(const float* __restrict__ A, const float* __restrict__ Wp,
                     const float* __restrict__ deg, const float* __restrict__ bias,
                     float* __restrict__ outP, int mode) {
    __shared__ float As[16 * DP];   // 13.3 KB A tile
    const int tid = threadIdx.x;
    const int rowbase = blockIdx.x * 16;

    // cooperative stage of 16 x 208 A tile into LDS
    for (int idx = tid; idx < 16 * DP; idx += 128) {
        int r = idx / DP, c = idx % DP;
        As[idx] = A[(long)(rowbase + r) * DP + c];
    }
    __syncthreads();

    const int wave = tid >> 5;
    const int lane = tid & 31;
    const int m    = lane & 15;      // A row within tile / B col within tile
    const int half = lane >> 4;      // K sub-slot (0 -> K+0,1 ; 1 -> K+2,3)

    for (int tile = wave; tile < 13; tile += 4) {
        const int col = tile * 16 + m;
        v8f acc = {0.f, 0.f, 0.f, 0.f, 0.f, 0.f, 0.f, 0.f};
        #pragma unroll 4
        for (int k0 = 0; k0 < DP; k0 += 4) {
            const int k = k0 + half * 2;
            // A frag (16x4 fp32): lane m holds A[m][k], A[m][k+1]
            float2 av = *reinterpret_cast<const float2*>(&As[m * DP + k]);
            v2f a; a.x = av.x; a.y = av.y;
            // B frag (4x16 fp32): lane holds W[k][col], W[k+1][col]
            v2f b; b.x = Wp[(long)k * DP + col];
            b.y = Wp[(long)(k + 1) * DP + col];
            acc = __builtin_amdgcn_wmma_f32_16x16x4_f32(
                false, a, false, b, (short)0, acc, false, false);
        }
        // epilogue: VGPR v holds rows v (lanes 0-15) / v+8 (lanes 16-31)
        #pragma unroll
        for (int v = 0; v < 8; ++v) {
            int row = rowbase + v + half * 8;
            float val = acc[v];
            if (mode == 0) {
                float invd = 1.0f / fmaxf(deg[row], 1.0f);
                val *= invd;
                val = (val >= 0.0f) ? val : SLOPE * val;
            } else {
                float bb = (col < DIM) ? bias[col] : 0.0f;
                val = 1.0f / (1.0f + expf(-(val + bb)));
            }
            outP[(long)row * DP + col] = (col < DIM) ? val : 0.0f;
        }
    }
}

// h_new = l2norm(gate*cur + (1-gate)*h) -> hP (padded) and out_t (dense)
__global__ void k_combine(const float* __restrict__ gateP, const float* __restrict__ curP,
                          float* __restrict__ hP, float* __restrict__ out_t, int N) {
    int wave = threadIdx.x >> 5, lane = threadIdx.x & 31;
    int row = blockIdx.x * 8 + wave;
    if (row >= N) return;
    float vals[7]; float ss = 0.0f;
    #pragma unroll
    for (int j = 0; j < 7; ++j) {
        int i = lane + j * 32;
        float v = 0.0f;
        if (i < DIM) {
            float g = gateP[(long)row * DP + i];
            float c = curP[(long)row * DP + i];
            float h = hP[(long)row * DP + i];
            v = g * c + (1.0f - g) * h;
        }
        vals[j] = v; ss += v * v;
    }
    ss = wave_sum32(ss);
    float inv = 1.0f / fmaxf(sqrtf(ss), 1e-12f);
    #pragma unroll
    for (int j = 0; j < 7; ++j) {
        int i = lane + j * 32;
        float v = vals[j] * inv;
        if (i < DP) hP[(long)row * DP + i] = (i < DIM) ? v : 0.0f;
        if (i < DIM) out_t[(long)row * DIM + i] = v;
    }
}

// ---------------------------------------------------------------------------

extern "C" void kernel_launch(void* const* d_in, const int* in_sizes, int n_in,
                              void* d_out, int out_size, void* d_ws, size_t ws_size,
                              hipStream_t stream) {
    const int T  = 8;
    const int E  = in_sizes[0] / T;          // 1,000,000
    const int N  = in_sizes[3] / DIM;        // 100,000
    const int R2 = in_sizes[4] / DIM;        // 460

    const int*   src  = (const int*)d_in[0];
    const int*   dst  = (const int*)d_in[1];
    const int*   et   = (const int*)d_in[2];
    const float* dyn  = (const float*)d_in[3];
    const float* rel  = (const float*)d_in[4];
    const float* W1   = (const float*)d_in[5];
    const float* W2   = (const float*)d_in[6];
    const float* Wg   = (const float*)d_in[7];
    const float* bias = (const float*)d_in[8];
    float* out = (float*)d_out;

    // workspace layout (floats)
    float* hP   = (float*)d_ws;              // N*DP
    float* preP = hP   + (long)N * DP;       // N*DP
    float* curP = preP + (long)N * DP;       // N*DP
    float* deg  = curP + (long)N * DP;       // N
    float* W1p  = deg  + N;                  // DP*DP
    float* W2p  = W1p  + DP * DP;
    float* Wgp  = W2p  + DP * DP;

    const long nPad = (long)N * DP;
    dim3 b256(256), b128(128);
    int rowBlocks = (N + 7) / 8;             // wave-per-row kernels
    int mmBlocks  = N / 16;                  // 6250
    int edgeWaveBlocks = (E + 7) / 8;        // wave-per-edge scatter

    // pad weights
    k_pad_weight<<<(DP * DP + 255) / 256, b256, 0, stream>>>(W1, W1p);
    k_pad_weight<<<(DP * DP + 255) / 256, b256, 0, stream>>>(W2, W2p);
    k_pad_weight<<<(DP * DP + 255) / 256, b256, 0, stream>>>(Wg, Wgp);

    // h0 = l2norm(dynamic_emb)
    k_init_h<<<rowBlocks, b256, 0, stream>>>(dyn, hP, N);

    for (int t = 0; t < T; ++t) {
        const int* st = src + (long)t * E;
        const int* dt = dst + (long)t * E;
        const int* rt = et  + (long)t * E;

        // in-degrees for this timestep
        k_zero<<<2048, b256, 0, stream>>>(deg, N);
        k_deg<<<(E + 255) / 256, b256, 0, stream>>>(dt, deg, E);

        // ---- layer 1: pre = scatter(h);  cur = rrelu((pre@W1)/deg)
        k_zero<<<4096, b256, 0, stream>>>(preP, nPad);
        k_scatter<<<edgeWaveBlocks, b256, 0, stream>>>(hP, st, dt, rt, rel, preP, E);
        k_mm<<<mmBlocks, b128, 0, stream>>>(preP, W1p, deg, nullptr, curP, 0);

        // ---- layer 2: pre = scatter(cur); cur = rrelu((pre@W2)/deg)
        k_zero<<<4096, b256, 0, stream>>>(preP, nPad);
        k_scatter<<<edgeWaveBlocks, b256, 0, stream>>>(curP, st, dt, rt, rel, preP, E);
        k_mm<<<mmBlocks, b128, 0, stream>>>(preP, W2p, deg, nullptr, curP, 0);

        // cur = l2norm(cur)
        k_l2norm_rows<<<rowBlocks, b256, 0, stream>>>(curP, N);

        // gate = sigmoid(h @ Wg + b)   (reuse preP as gate buffer)
        k_mm<<<mmBlocks, b128, 0, stream>>>(hP, Wgp, nullptr, bias, preP, 1);

        // h = l2norm(gate*cur + (1-gate)*h); emit evolve[t]
        k_combine<<<rowBlocks, b256, 0, stream>>>(preP, curP, hP,
                                                  out + (long)t * N * DIM, N);
    }

    // tuple output: (evolve, emb_rel) concatenated
    k_copy<<<(R2 * DIM + 255) / 256, b256, 0, stream>>>(
        rel, out + (long)T * N * DIM, R2 * DIM);
}